// YOLOLoss_67577015435969
// MI455X (gfx1250) — compile-verified
//
#include <hip/hip_runtime.h>
#include <math.h>

typedef float v2f __attribute__((ext_vector_type(2)));
typedef float v8f __attribute__((ext_vector_type(8)));

#define NUM_CLASSES 80
#define T_PER_B 20
#define BATCH 32
#define LOG2_CONST 0.69314718055994530942f

// ws layout (floats)
#define WS_FULL  0     // [0..95]    full[s][b] = sum softplus(obj plane)
#define WS_BOX   96    // [96..98]   box[s]
#define WS_OBJ   99    // [99..101]  obj[s]
#define WS_SPOBJ 102   // [102..104] sum vm*softplus(pobj) [s]
#define WS_CLS   105   // [105..107] cls[s]
#define WS_NV    108   // [108..203] nv[s][b]
#define WS_TOTAL 204

__device__ __forceinline__ float softplus_f(float x) {
    return fmaxf(x, 0.0f) + log1pf(expf(-fabsf(x)));
}
__device__ __forceinline__ float sigmoid_f(float x) {
    return 1.0f / (1.0f + expf(-x));
}

__global__ void zero_ws(float* ws) {
    int i = threadIdx.x;
    if (i < WS_TOTAL) ws[i] = 0.0f;
}

// One block per (scale, batch, anchor) objectness plane.
// Block-reduces sum(softplus(plane)) using chained f32 WMMAs (A = ones trick).
__global__ void obj_plane_kernel(const float* __restrict__ p0,
                                 const float* __restrict__ p1,
                                 const float* __restrict__ p2,
                                 float* __restrict__ ws) {
    int blk = blockIdx.x;            // 0..287
    int s   = blk / 96;
    int rem = blk % 96;
    int b   = rem / 3;
    int a   = rem % 3;
    int G   = (s == 0) ? 13 : (s == 1) ? 26 : 52;
    int GG  = G * G;
    const float* p = (s == 0) ? p0 : (s == 1) ? p1 : p2;
    const float* plane = p + ((size_t)b * 255 + a * 85 + 4) * (size_t)GG;

    float acc = 0.0f;
    for (int i = threadIdx.x; i < GG; i += blockDim.x)
        acc += softplus_f(plane[i]);

    __shared__ float red[256];
    red[threadIdx.x] = acc;
    __syncthreads();

    if (threadIdx.x < 32) {          // wave 0, EXEC all ones for WMMA
        int lane = threadIdx.x;
        float v[8];
        #pragma unroll
        for (int i = 0; i < 8; ++i) v[i] = red[i * 32 + lane];

        v2f ones; ones[0] = 1.0f; ones[1] = 1.0f;   // A = 16x4 all-ones
        v8f c = {};
        #pragma unroll
        for (int k = 0; k < 4; ++k) {
            v2f bm; bm[0] = v[2 * k]; bm[1] = v[2 * k + 1];  // 64 values as B (4x16)
            // D[m,n] += sum_k B[k,n]; columns partition the 64 values exactly once.
            c = __builtin_amdgcn_wmma_f32_16x16x4_f32(
                    false, ones, false, bm, (short)0, c, false, false);
        }
        // C/D layout: VGPR0 lanes 0..15 = D[M=0][N=lane] = column sums S_n
        // (lanes 16..31 hold the identical row M=8). Fold 16 columns:
        float colsum = c[0];
        #pragma unroll
        for (int off = 8; off > 0; off >>= 1)
            colsum += __shfl_xor(colsum, off, 32);
        if (lane == 0)
            atomicAdd(&ws[WS_FULL + s * 32 + b], colsum);
    }
}

// One wave per batch; lane = target index (lanes >= 20 contribute zero).
__global__ void target_kernel(const float* __restrict__ p0,
                              const float* __restrict__ p1,
                              const float* __restrict__ p2,
                              const float* __restrict__ targets,
                              const float* __restrict__ anchors,
                              float* __restrict__ ws) {
    int b = blockIdx.x;
    int t = threadIdx.x;

    float cls_f = 0.0f, x1 = 0.0f, y1 = 0.0f, x2 = 0.0f, y2 = 0.0f;
    bool have = (t < T_PER_B);
    if (have) {
        const float* tg = targets + ((size_t)b * T_PER_B + t) * 5;
        cls_f = tg[0]; x1 = tg[1]; y1 = tg[2]; x2 = tg[3]; y2 = tg[4];
    }

    for (int s = 0; s < 3; ++s) {
        int G = (s == 0) ? 13 : (s == 1) ? 26 : 52;
        float stride = (s == 0) ? 32.0f : (s == 1) ? 16.0f : 8.0f;
        const float* p = (s == 0) ? p0 : (s == 1) ? p1 : p2;
        size_t GG = (size_t)G * G;

        float box_t = 0.0f, obj_t = 0.0f, spobj_t = 0.0f, cls_t = 0.0f, nv_t = 0.0f;
        if (have) {
            float gx = x1 / stride, gy = y1 / stride;
            float gw = (x2 - x1) / stride, gh = (y2 - y1) / stride;
            int gi = (int)floorf(gx), gj = (int)floorf(gy);
            bool valid = (gi >= 0) && (gi < G) && (gj >= 0) && (gj < G);
            int gic = min(max(gi, 0), G - 1);
            int gjc = min(max(gj, 0), G - 1);

            int best = 0; float bestv = -1.0f, aw_best = 0.0f, ah_best = 0.0f;
            #pragma unroll
            for (int a = 0; a < 3; ++a) {
                float aw = anchors[(s * 3 + a) * 2 + 0] / stride;
                float ah = anchors[(s * 3 + a) * 2 + 1] / stride;
                float inter = fminf(gw, aw) * fminf(gh, ah);
                float uni   = gw * gh + aw * ah - inter;
                float iou   = inter / (uni + 1e-16f);
                if (iou > bestv) { bestv = iou; best = a; aw_best = aw; ah_best = ah; }
            }

            float gt_x = gx - (float)gic;
            float gt_y = gy - (float)gjc;
            float gt_w = logf(fmaxf(gw, 1e-16f) / (aw_best + 1e-16f));
            float gt_h = logf(fmaxf(gh, 1e-16f) / (ah_best + 1e-16f));

            const float* base = p + ((size_t)b * 255 + best * 85) * GG
                                  + (size_t)gjc * G + gic;
            float px   = sigmoid_f(base[0]);
            float py   = sigmoid_f(base[GG]);
            float pw   = base[2 * GG];
            float ph   = base[3 * GG];
            float pobj = base[4 * GG];

            float dx = px - gt_x, dy = py - gt_y, dw = pw - gt_w, dh = ph - gt_h;
            box_t   = dx * dx + dy * dy + dw * dw + dh * dh;
            obj_t   = softplus_f(-pobj);
            spobj_t = softplus_f(pobj);

            int tcls = min(max((int)cls_f, 0), NUM_CLASSES - 1);
            float csum = 0.0f;
            for (int c = 0; c < NUM_CLASSES; ++c)
                csum += softplus_f(base[(size_t)(5 + c) * GG]);
            csum -= base[(size_t)(5 + tcls) * GG];
            cls_t = csum;

            float vm = valid ? 1.0f : 0.0f;
            box_t *= vm; obj_t *= vm; spobj_t *= vm; cls_t *= vm; nv_t = vm;
        }

        for (int off = 16; off > 0; off >>= 1) {
            box_t   += __shfl_xor(box_t,   off, 32);
            obj_t   += __shfl_xor(obj_t,   off, 32);
            spobj_t += __shfl_xor(spobj_t, off, 32);
            cls_t   += __shfl_xor(cls_t,   off, 32);
            nv_t    += __shfl_xor(nv_t,    off, 32);
        }
        if (t == 0) {
            atomicAdd(&ws[WS_BOX   + s], box_t);
            atomicAdd(&ws[WS_OBJ   + s], obj_t);
            atomicAdd(&ws[WS_SPOBJ + s], spobj_t);
            atomicAdd(&ws[WS_CLS   + s], cls_t);
            ws[WS_NV + s * 32 + b] = nv_t;   // sole writer for (s,b)
        }
    }
}

__global__ void finalize_kernel(const float* __restrict__ ws,
                                float* __restrict__ out) {
    if (threadIdx.x == 0 && blockIdx.x == 0) {
        float total = 0.0f;
        for (int s = 0; s < 3; ++s) {
            float nvfull = 0.0f, sumnv = 0.0f;
            for (int b = 0; b < BATCH; ++b) {
                float nv = ws[WS_NV + s * 32 + b];
                nvfull += nv * ws[WS_FULL + s * 32 + b];
                sumnv  += nv;
            }
            float noobj = nvfull - ws[WS_SPOBJ + s] + sumnv * (float)LOG2_CONST;
            total += 5.0f * ws[WS_BOX + s] + ws[WS_OBJ + s]
                   + 50.0f * noobj + ws[WS_CLS + s];
        }
        out[0] = total / (float)BATCH;
    }
}

extern "C" void kernel_launch(void* const* d_in, const int* in_sizes, int n_in,
                              void* d_out, int out_size, void* d_ws, size_t ws_size,
                              hipStream_t stream) {
    const float* p0      = (const float*)d_in[0];
    const float* p1      = (const float*)d_in[1];
    const float* p2      = (const float*)d_in[2];
    const float* targets = (const float*)d_in[3];
    const float* anchors = (const float*)d_in[4];
    float* ws  = (float*)d_ws;
    float* out = (float*)d_out;

    zero_ws<<<1, 256, 0, stream>>>(ws);
    obj_plane_kernel<<<288, 256, 0, stream>>>(p0, p1, p2, ws);
    target_kernel<<<32, 32, 0, stream>>>(p0, p1, p2, targets, anchors, ws);
    finalize_kernel<<<1, 32, 0, stream>>>(ws, out);
}